// DDLMS_16535624089792
// MI455X (gfx1250) — compile-verified
//
#include <hip/hip_runtime.h>
#include <math.h>

#define TAPS       32
#define PREF_DEPTH 16           // LDS ring depth (power of 2); 16*512B = 8KB

#define LR_W   0.0625f          // 1/2^4
#define LR_F   0.0078125f       // 1/2^7
#define LR_B   0.00048828125f   // 1/2^11
#define GMAX   30.0f
#define EPSF   1e-8f
#define INV_S10 0.31622776601683794f   // 1/sqrt(10)
#define TH_S10  0.6324555320336759f    // 2/sqrt(10)

// ---- wave32 all-lanes sum via DPP16 rotate-and-add + permlanex16 (no LDS) ----
template <int CTRL>
__device__ __forceinline__ float row_ror_add(float x) {
  int xi = __float_as_int(x);
  int yi = __builtin_amdgcn_update_dpp(xi, xi, CTRL, 0xf, 0xf, false);
  return x + __int_as_float(yi);
}

__device__ __forceinline__ float wsum(float x) {
  // within each 16-lane row: rotate-reduce (every lane ends with the row sum)
  x = row_ror_add<0x121>(x);   // row_ror:1
  x = row_ror_add<0x122>(x);   // row_ror:2
  x = row_ror_add<0x124>(x);   // row_ror:4
  x = row_ror_add<0x128>(x);   // row_ror:8
  // cross the two 16-lane halves: identity-select permlanex16 == lane^16 swap
  int xi = __float_as_int(x);
  int yi = __builtin_amdgcn_permlanex16(xi, xi, 0x76543210, 0xFEDCBA98, false, false);
  return x + __int_as_float(yi);
}

// QAM16 square-grid slicer (== argmin over the unit-power constellation)
__device__ __forceinline__ float slice1d(float x) {
  return copysignf(fabsf(x) < TH_S10 ? INV_S10 : 3.0f * INV_S10, x);
}

__global__ __launch_bounds__(32, 1)
void ddlms_scan(const float* __restrict__ uin, float* __restrict__ out, int nsteps) {
  // ring of u-tiles; one slot = 32 taps * (2 dims * re/im) = 128 floats = 512B
  __shared__ __align__(16) float ubuf[PREF_DEPTH][TAPS * 4];
  const int lane = threadIdx.x & 31;
  if (nsteps <= 0) return;

  // LDS byte offset of ubuf: low 32 bits of the flat address are the LDS offset
  const unsigned lds0 = (unsigned)(uintptr_t)(void*)&ubuf[0][0];

  // -------- prologue: fill the async ring (ASYNCcnt-tracked, in-order) --------
  for (int i = 0; i < PREF_DEPTH; ++i) {
    int pn = (i < nsteps) ? i : (nsteps - 1);
    const float* g = uin + (size_t)pn * (TAPS * 4) + lane * 4;
    unsigned la = lds0 + (unsigned)(i * (TAPS * 4 * 4) + lane * 16);
    asm volatile("global_load_async_to_lds_b128 %0, %1, off"
                 :: "v"(la), "v"(g) : "memory");
  }

  // per-lane filter state: w[i][j][t=lane], i,j in {0,1}
  float w00r = 0.f, w00i = 0.f, w01r = 0.f, w01i = 0.f;
  float w10r = 0.f, w10i = 0.f, w11r = 0.f, w11i = 0.f;
  // uniform state (redundant in all lanes). s==1 (LR_S==0); fshat==f*s==f (BETA==0) => q==z
  float f0r = 1.f, f0i = 0.f, f1r = 1.f, f1i = 0.f;
  float b0r = 0.f, b0i = 0.f, b1r = 0.f, b1i = 0.f;

  for (int n = 0; n < nsteps; ++n) {
    const int slot = n & (PREF_DEPTH - 1);

    // wait until the slot for step n has landed (in-order completion => cnt<=DEPTH-1 suffices)
    asm volatile("s_wait_asynccnt %0" :: "n"(PREF_DEPTH - 1) : "memory");
    const float4 uv = *(const float4*)&ubuf[slot][lane * 4]; // (u0r,u0i,u1r,u1i) for tap=lane

    // per-lane partial MACs: v[i] = sum_{t,j} w[i][j][t]*u[t][j]
    float p0r = w00r * uv.x - w00i * uv.y + w01r * uv.z - w01i * uv.w;
    float p0i = w00r * uv.y + w00i * uv.x + w01r * uv.w + w01i * uv.z;
    float p1r = w10r * uv.x - w10i * uv.y + w11r * uv.z - w11i * uv.w;
    float p1i = w10r * uv.y + w10i * uv.x + w11r * uv.w + w11i * uv.z;
    float puu = uv.x * uv.x + uv.y * uv.y + uv.z * uv.z + uv.w * uv.w;

    // refill this (now dead) slot with u[n+PREF_DEPTH]
    {
      int pn = n + PREF_DEPTH;
      if (pn >= nsteps) pn = nsteps - 1;          // tail reloads never read again
      const float* g = uin + (size_t)pn * (TAPS * 4) + lane * 4;
      unsigned la = lds0 + (unsigned)(slot * (TAPS * 4 * 4) + lane * 16);
      asm volatile("global_load_async_to_lds_b128 %0, %1, off"
                   :: "v"(la), "v"(g) : "memory");
    }

    // wave reduction -> every lane holds v0, v1, uu (pure VALU: DPP + permlanex16)
    const float v0r = wsum(p0r), v0i = wsum(p0i);
    const float v1r = wsum(p1r), v1i = wsum(p1i);
    const float uu  = wsum(puu);

    // ---- uniform scalar section (s==1, fshat==f => q==z, e_f==e_s==e_b=d-z) ----
    const float k0r = v0r * f0r - v0i * f0i, k0i = v0r * f0i + v0i * f0r;
    const float k1r = v1r * f1r - v1i * f1i, k1i = v1r * f1i + v1i * f1r;
    const float z0r = k0r + b0r, z0i = k0i + b0i;
    const float z1r = k1r + b1r, z1i = k1i + b1i;

    const float d0r = slice1d(z0r), d0i = slice1d(z0i);
    const float d1r = slice1d(z1r), d1i = slice1d(z1i);

    const float e0r = d0r - z0r, e0i = d0i - z0i;
    const float e1r = d1r - z1r, e1i = d1i - z1i;

    // psi = conj(f)/|f|   (|s|/s == 1)
    const float pf0 = __builtin_amdgcn_rsqf(f0r * f0r + f0i * f0i);
    const float pf1 = __builtin_amdgcn_rsqf(f1r * f1r + f1i * f1i);
    const float ps0r = f0r * pf0, ps0i = -f0i * pf0;
    const float ps1r = f1r * pf1, ps1i = -f1i * pf1;

    // e_w = (d - b)*psi - v
    const float db0r = d0r - b0r, db0i = d0i - b0i;
    const float db1r = d1r - b1r, db1i = d1i - b1i;
    const float ew0r = db0r * ps0r - db0i * ps0i - v0r;
    const float ew0i = db0r * ps0i + db0i * ps0r - v0i;
    const float ew1r = db1r * ps1r - db1i * ps1i - v1r;
    const float ew1i = db1r * ps1i + db1i * ps1r - v1i;

    // f update: gf = -e*conj(v)/(|v|^2+eps); clip |gf|<=30; f -= LR_F*gf
    {
      const float t0 = -__builtin_amdgcn_rcpf(v0r * v0r + v0i * v0i + EPSF);
      const float t1 = -__builtin_amdgcn_rcpf(v1r * v1r + v1i * v1i + EPSF);
      const float g0r = t0 * (e0r * v0r + e0i * v0i), g0i = t0 * (e0i * v0r - e0r * v0i);
      const float g1r = t1 * (e1r * v1r + e1i * v1i), g1i = t1 * (e1i * v1r - e1r * v1i);
      const float c0 = fminf(1.f, GMAX * __builtin_amdgcn_rsqf(g0r * g0r + g0i * g0i));
      const float c1 = fminf(1.f, GMAX * __builtin_amdgcn_rsqf(g1r * g1r + g1i * g1i));
      f0r -= LR_F * c0 * g0r; f0i -= LR_F * c0 * g0i;
      f1r -= LR_F * c1 * g1r; f1i -= LR_F * c1 * g1i;
    }

    // b update: gb = -e_b => b += LR_B*e
    b0r += LR_B * e0r; b0i += LR_B * e0i;
    b1r += LR_B * e1r; b1i += LR_B * e1i;

    // w update (per-lane, 4 owned complex taps): g = cw*e_w[i]*conj(u_j); clip; w -= LR_W*g
    {
      const float cw = -__builtin_amdgcn_rcpf(uu + EPSF);
      float gr, gi, m;
      gr = cw * (ew0r * uv.x + ew0i * uv.y); gi = cw * (ew0i * uv.x - ew0r * uv.y);
      m  = fminf(1.f, GMAX * __builtin_amdgcn_rsqf(gr * gr + gi * gi));
      w00r -= LR_W * m * gr; w00i -= LR_W * m * gi;
      gr = cw * (ew0r * uv.z + ew0i * uv.w); gi = cw * (ew0i * uv.z - ew0r * uv.w);
      m  = fminf(1.f, GMAX * __builtin_amdgcn_rsqf(gr * gr + gi * gi));
      w01r -= LR_W * m * gr; w01i -= LR_W * m * gi;
      gr = cw * (ew1r * uv.x + ew1i * uv.y); gi = cw * (ew1i * uv.x - ew1r * uv.y);
      m  = fminf(1.f, GMAX * __builtin_amdgcn_rsqf(gr * gr + gi * gi));
      w10r -= LR_W * m * gr; w10i -= LR_W * m * gi;
      gr = cw * (ew1r * uv.z + ew1i * uv.w); gi = cw * (ew1i * uv.z - ew1r * uv.w);
      m  = fminf(1.f, GMAX * __builtin_amdgcn_rsqf(gr * gr + gi * gi));
      w11r -= LR_W * m * gr; w11i -= LR_W * m * gi;
    }

    // emit signal[n] = z (weights-before-update path): [n][dim][re,im]
    if (lane == 0) {
      float4 o; o.x = z0r; o.y = z0i; o.z = z1r; o.w = z1i;
      *(float4*)(out + (size_t)n * 4) = o;
    }
  }
}

extern "C" void kernel_launch(void* const* d_in, const int* in_sizes, int n_in,
                              void* d_out, int out_size, void* d_ws, size_t ws_size,
                              hipStream_t stream) {
  const float* u = (const float*)d_in[0];
  float* out = (float*)d_out;
  const int nsteps = in_sizes[0] / (TAPS * 2 * 2);   // [N, TAPS, DIMS, 2] f32
  hipLaunchKernelGGL(ddlms_scan, dim3(1), dim3(32), 0, stream, u, out, nsteps);
}